// EglGINConv_70978629534133
// MI455X (gfx1250) — compile-verified
//
#include <hip/hip_runtime.h>
#include <hip/hip_bf16.h>

typedef __attribute__((ext_vector_type(2))) float v2f;
typedef __attribute__((ext_vector_type(8))) float v8f;

#define N_NODES 100000
#define N_EDGES 1600000
#define DIM 64

// ---------------------------------------------------------------------------
// Kernel 1: rst[i] = (1+eps) * feat[i]   (float4 vectorized)
// ---------------------------------------------------------------------------
__global__ __launch_bounds__(256) void gin_init_kernel(
    const float* __restrict__ feat, const float* __restrict__ eps,
    float* __restrict__ rst, int n4) {
  int i = blockIdx.x * blockDim.x + threadIdx.x;
  if (i >= n4) return;
  float s = 1.0f + eps[0];
  float4 f = ((const float4*)feat)[i];
  f.x *= s; f.y *= s; f.z *= s; f.w *= s;
  ((float4*)rst)[i] = f;
}

// ---------------------------------------------------------------------------
// Kernel 2: scatter-add  rst[dst[e]] += feat[src[e]]
// 32 lanes per edge; each lane handles 2 floats (float2 gather + 2 f32 atomics)
// ---------------------------------------------------------------------------
__global__ __launch_bounds__(256) void gin_scatter_kernel(
    const float* __restrict__ feat, const int* __restrict__ src,
    const int* __restrict__ dst, float* __restrict__ rst, int nEdges) {
  int gid = blockIdx.x * blockDim.x + threadIdx.x;
  int e = gid >> 5;
  if (e >= nEdges) return;
  int lane = gid & 31;
  int s = src[e];
  int d = dst[e];
  const float2 v = *(const float2*)(feat + (size_t)s * DIM + lane * 2);
  float* p = rst + (size_t)d * DIM + lane * 2;
  unsafeAtomicAdd(p, v.x);       // -> global_atomic_add_f32 (no return)
  unsafeAtomicAdd(p + 1, v.y);
}

// ---------------------------------------------------------------------------
// Kernel 3: out = rst @ W + b   via V_WMMA_F32_16X16X4_F32
// Block: 256 threads = 8 waves. Block tile: 32 rows x 64 cols.
// Each wave: one 16x16 output tile, K=64 in 16 WMMA steps.
// LDS: A tile (32x64, row stride 68 floats) + W^T (64x64, row stride 68).
// Stride 68 floats = 272 B: 16B-aligned rows, bank stride 4 -> conflict-free.
// ---------------------------------------------------------------------------
__global__ __launch_bounds__(256) void gin_gemm_kernel(
    const float* __restrict__ rst, const float* __restrict__ W,
    const float* __restrict__ bias, float* __restrict__ out) {
  __shared__ float sA[32 * 68];    //  8704 B
  __shared__ float sWt[64 * 68];   // 17408 B  (sWt[c*68 + k] = W[k][c])

  const int tid  = threadIdx.x;
  const int lane = tid & 31;
  const int wave = tid >> 5;
  const int rowBase = blockIdx.x * 32;

  // Stage W transposed into LDS (4096 elements, 16 per thread).
  for (int i = tid; i < DIM * DIM; i += 256) {
    int k = i >> 6;      // row of W
    int c = i & 63;      // col of W
    sWt[c * 68 + k] = W[i];
  }
  // Stage A tile: 32 rows x 64 cols = 512 float4 loads, 2 per thread.
  for (int i = tid; i < 512; i += 256) {
    int r  = i >> 4;          // 0..31
    int c4 = (i & 15) * 4;    // 0,4,...,60
    float4 v = *(const float4*)(rst + (size_t)(rowBase + r) * DIM + c4);
    *(float4*)&sA[r * 68 + c4] = v;  // 272B row stride keeps 16B alignment
  }
  __syncthreads();

  const int mTile = wave >> 2;        // 0..1 : which 16-row half
  const int nTile = wave & 3;         // 0..3 : which 16-col slice
  const int rlo   = lane & 15;        // row/col within tile
  const int hi    = lane >> 4;        // K-half selector (ISA A/B layout)

  const int aRow = mTile * 16 + rlo;          // row in sA
  const int bCol = nTile * 16 + rlo;          // column of W (row of sWt)

  v8f acc = {};
#pragma unroll
  for (int kt = 0; kt < 16; ++kt) {
    const int k0 = kt * 4 + hi * 2;
    // A fragment: lane holds rst[row, k0], rst[row, k0+1]
    v2f a = *(const v2f*)&sA[aRow * 68 + k0];
    // B fragment: lane holds W[k0][col], W[k0+1][col]  (contiguous in sWt)
    v2f b = *(const v2f*)&sWt[bCol * 68 + k0];
    acc = __builtin_amdgcn_wmma_f32_16x16x4_f32(
        /*neg_a=*/false, a, /*neg_b=*/false, b,
        /*c_mod=*/(short)0, acc, /*reuse_a=*/false, /*reuse_b=*/false);
  }

  // C/D layout: VGPR v -> row M = v + 8*hi (lanes 0-15: M=v; 16-31: M=v+8)
  const int col  = nTile * 16 + rlo;
  const float bb = bias[col];
  const int orow = rowBase + mTile * 16 + hi * 8;
#pragma unroll
  for (int v = 0; v < 8; ++v) {
    out[(size_t)(orow + v) * DIM + col] = acc[v] + bb;
  }
}

// ---------------------------------------------------------------------------
extern "C" void kernel_launch(void* const* d_in, const int* in_sizes, int n_in,
                              void* d_out, int out_size, void* d_ws, size_t ws_size,
                              hipStream_t stream) {
  const float* feat = (const float*)d_in[0];
  const int*   src  = (const int*)d_in[1];
  const int*   dst  = (const int*)d_in[2];
  const float* eps  = (const float*)d_in[3];
  const float* W    = (const float*)d_in[4];
  const float* bias = (const float*)d_in[5];
  float* out = (float*)d_out;
  float* rst = (float*)d_ws;   // N_NODES * DIM floats = 25.6 MB scratch

  // 1) rst = (1+eps)*feat
  {
    int n4 = N_NODES * DIM / 4;                      // 1.6M float4
    gin_init_kernel<<<(n4 + 255) / 256, 256, 0, stream>>>(feat, eps, rst, n4);
  }
  // 2) rst[dst] += feat[src]  (32 lanes per edge)
  {
    long long threads = (long long)N_EDGES * 32;     // 51.2M
    int blocks = (int)((threads + 255) / 256);       // 200000
    gin_scatter_kernel<<<blocks, 256, 0, stream>>>(feat, src, dst, rst, N_EDGES);
  }
  // 3) out = rst @ W + b  (WMMA f32 16x16x4)
  {
    int blocks = N_NODES / 32;                       // 3125, exact
    gin_gemm_kernel<<<blocks, 256, 0, stream>>>(rst, W, bias, out);
  }
}